// MessagePassingCouplingLayer_7627861918011
// MI455X (gfx1250) — compile-verified
//
#include <hip/hip_runtime.h>

#define B_   32
#define N_   8192
#define E_   1048576
#define H_   128
#define EMB_ 64
#define L_   2

typedef float v2f __attribute__((ext_vector_type(2)));
typedef float v8f __attribute__((ext_vector_type(8)));
typedef int   v4i __attribute__((ext_vector_type(4)));

// D = A(16x4,f32) * B(4x16,f32) + C(16x16,f32)  — full-precision matrix-pipe op
__device__ __forceinline__ v8f wmma_f32(v2f a, v2f b, v8f c) {
  return __builtin_amdgcn_wmma_f32_16x16x4_f32(false, a, false, b, (short)0, c, false, false);
}

// ---- CDNA5 async global->LDS path (ASYNCcnt), with safe fallback ----------
#if defined(__has_builtin)
#if __has_builtin(__builtin_amdgcn_global_load_async_to_lds_b128)
#define HAVE_ASYNC_LDS 1
#endif
#endif

#ifdef HAVE_ASYNC_LDS
typedef __attribute__((address_space(1))) v4i* gptr_v4i;
typedef __attribute__((address_space(3))) v4i* lptr_v4i;

__device__ __forceinline__ void async_copy16(const float* g, float* l) {
  v4i* gg = (v4i*)g;                    // reinterpret (drops const), generic AS
  v4i* ll = (v4i*)l;
  __builtin_amdgcn_global_load_async_to_lds_b128((gptr_v4i)gg, (lptr_v4i)ll, 0, 0);
}
__device__ __forceinline__ void wait_async0() {
#if __has_builtin(__builtin_amdgcn_s_wait_asynccnt)
  __builtin_amdgcn_s_wait_asynccnt(0);
#else
  asm volatile("s_wait_asynccnt 0x0" ::: "memory");
#endif
}
#endif

// ---------------------------------------------------------------------------
// Kernel 1: h = relu([embed[type] , coords] @ W_in + b_in) * valid
// One block = 16 nodes x 128 cols; 8 waves, each wave owns 16 output columns.
// ---------------------------------------------------------------------------
__global__ __launch_bounds__(256)
void input_layer_kernel(const float* __restrict__ coords,
                        const int* __restrict__ atypes,
                        const unsigned char* __restrict__ maskel,
                        const float* __restrict__ embed,
                        const float* __restrict__ W_in,
                        const float* __restrict__ b_in,
                        float* __restrict__ h)
{
  const int KP = 68, LD = 69;            // K=67 padded to 68, odd LDS stride
  __shared__ float feat[16 * LD];
  __shared__ float validf[16];
  const int tid = threadIdx.x;
  const long base = (long)blockIdx.x * 16;

  if (tid < 16) validf[tid] = maskel[base + tid] ? 0.f : 1.f;
  for (int idx = tid; idx < 16 * KP; idx += 256) {
    int r = idx / KP, c = idx % KP;
    long node = base + r;
    float v = 0.f;
    if (c < EMB_)            v = embed[(long)atypes[node] * EMB_ + c];
    else if (c < EMB_ + 3)   v = coords[node * 3 + (c - EMB_)];
    feat[r * LD + c] = v;
  }
  __syncthreads();

  const int lane = tid & 31, wave = tid >> 5;
  const int col = lane & 15;             // N (and A-row M) index
  const int khalf = (lane >> 4) * 2;     // K sub-offset per lane half
  const int n0 = wave * 16;

  v8f acc;
  const float bv = b_in[n0 + col];
  #pragma unroll
  for (int j = 0; j < 8; ++j) acc[j] = bv;

  #pragma unroll
  for (int s = 0; s < 17; ++s) {
    int k0 = 4 * s + khalf;
    v2f a; a.x = feat[col * LD + k0]; a.y = feat[col * LD + k0 + 1];
    v2f b;
    b.x = (k0     < 67) ? W_in[(k0    ) * H_ + n0 + col] : 0.f;
    b.y = (k0 + 1 < 67) ? W_in[(k0 + 1) * H_ + n0 + col] : 0.f;
    acc = wmma_f32(a, b, acc);
  }

  const int rbase = (lane >> 4) * 8;
  #pragma unroll
  for (int j = 0; j < 8; ++j) {
    int r = rbase + j;
    h[(base + r) * H_ + n0 + col] = fmaxf(acc[j], 0.f) * validf[r];
  }
}

// ---------------------------------------------------------------------------
// Kernel 2: messages: m = relu(h[src] @ msg_W + msg_b); agg[dst] += m
// Grid-stride over 16-edge tiles; each wave keeps its 16-column B panel
// (32 x v2f = 64 VGPRs) resident across tiles. The h[src] row gather — the
// dominant HBM traffic — goes over the CDNA5 async global->LDS path: one
// GLOBAL_LOAD_ASYNC_TO_LDS_B128 per lane moves a full 512B row per wave.
// ---------------------------------------------------------------------------
__global__ __launch_bounds__(256)
void message_kernel(const float* __restrict__ h,
                    const int* __restrict__ adj,
                    const int* __restrict__ ebi,
                    const float* __restrict__ W,     // [128,128]
                    const float* __restrict__ bias,  // [128]
                    float* __restrict__ agg,
                    int ntiles)
{
  const int LD = 132;                    // x4B = 528: 16B-aligned rows, banks 4r+k
  __shared__ float hs[16 * LD];
  __shared__ int ssrc[16];
  __shared__ int sdst[16];

  const int tid = threadIdx.x;
  const int lane = tid & 31, wave = tid >> 5;
  const int col = lane & 15;
  const int khalf = (lane >> 4) * 2;
  const int n0 = wave * 16;

  v2f bp[32];
  #pragma unroll
  for (int s = 0; s < 32; ++s) {
    int k0 = 4 * s + khalf;
    bp[s].x = W[(k0    ) * H_ + n0 + col];
    bp[s].y = W[(k0 + 1) * H_ + n0 + col];
  }
  const float bv = bias[n0 + col];
  const int rbase = (lane >> 4) * 8;

  for (int tile = blockIdx.x; tile < ntiles; tile += gridDim.x) {
    __syncthreads();                     // protect LDS reuse across iterations
    if (tid < 16) {
      long e = (long)tile * 16 + tid;
      int b = ebi[e];
      ssrc[tid] = b * N_ + adj[2 * e];
      sdst[tid] = b * N_ + adj[2 * e + 1];
    }
    __syncthreads();
#ifdef HAVE_ASYNC_LDS
    {
      int r0 = wave * 2;                 // each wave stages 2 rows
      const float* g0 = h + (long)ssrc[r0    ] * H_ + lane * 4;
      const float* g1 = h + (long)ssrc[r0 + 1] * H_ + lane * 4;
      async_copy16(g0, &hs[(r0    ) * LD + lane * 4]);
      async_copy16(g1, &hs[(r0 + 1) * LD + lane * 4]);
      wait_async0();
    }
#else
    for (int idx = tid; idx < 16 * H_; idx += 256) {
      int r = idx >> 7, c = idx & 127;
      hs[r * LD + c] = h[(long)ssrc[r] * H_ + c];
    }
#endif
    __syncthreads();

    v8f acc;
    #pragma unroll
    for (int j = 0; j < 8; ++j) acc[j] = bv;
    #pragma unroll
    for (int s = 0; s < 32; ++s) {
      int k0 = 4 * s + khalf;
      v2f a; a.x = hs[col * LD + k0]; a.y = hs[col * LD + k0 + 1];
      acc = wmma_f32(a, bp[s], acc);
    }
    #pragma unroll
    for (int j = 0; j < 8; ++j) {
      float v = fmaxf(acc[j], 0.f);
      atomicAdd(&agg[(long)sdst[rbase + j] * H_ + n0 + col], v);
    }
  }
}

// ---------------------------------------------------------------------------
// Kernel 3: h = relu([h , agg] @ upd_W + upd_b) * valid   (in-place, K=256)
// ---------------------------------------------------------------------------
__global__ __launch_bounds__(256)
void update_kernel(float* __restrict__ h,
                   const float* __restrict__ agg,
                   const float* __restrict__ W,     // [256,128]
                   const float* __restrict__ bias,
                   const unsigned char* __restrict__ maskel)
{
  const int LD = 260;                    // mult of 4 (16B rows), banks 4r+k
  __shared__ float fs[16 * LD];
  __shared__ float validf[16];
  const int tid = threadIdx.x;
  const int lane = tid & 31, wave = tid >> 5;
  const long base = (long)blockIdx.x * 16;

  if (tid < 16) validf[tid] = maskel[base + tid] ? 0.f : 1.f;
#ifdef HAVE_ASYNC_LDS
  {
    int r0 = wave * 2;
    #pragma unroll
    for (int i = 0; i < 2; ++i) {
      int r = r0 + i;
      async_copy16(h   + (base + r) * H_ + lane * 4, &fs[r * LD +       lane * 4]);
      async_copy16(agg + (base + r) * H_ + lane * 4, &fs[r * LD + H_ +  lane * 4]);
    }
    wait_async0();
  }
#else
  for (int idx = tid; idx < 16 * H_; idx += 256) {
    int r = idx >> 7, c = idx & 127;
    fs[r * LD + c]      = h  [(base + r) * H_ + c];
    fs[r * LD + H_ + c] = agg[(base + r) * H_ + c];
  }
#endif
  __syncthreads();

  const int col = lane & 15, khalf = (lane >> 4) * 2, n0 = wave * 16;

  v8f acc;
  const float bv = bias[n0 + col];
  #pragma unroll
  for (int j = 0; j < 8; ++j) acc[j] = bv;
  #pragma unroll
  for (int s = 0; s < 64; ++s) {
    int k0 = 4 * s + khalf;
    v2f a; a.x = fs[col * LD + k0]; a.y = fs[col * LD + k0 + 1];
    v2f b; b.x = W[(k0) * H_ + n0 + col]; b.y = W[(k0 + 1) * H_ + n0 + col];
    acc = wmma_f32(a, b, acc);
  }
  const int rbase = (lane >> 4) * 8;
  #pragma unroll
  for (int j = 0; j < 8; ++j) {
    int r = rbase + j;
    h[(base + r) * H_ + n0 + col] = fmaxf(acc[j], 0.f) * validf[r];
  }
}

// ---------------------------------------------------------------------------
// Kernel 4: node_features = h @ W_out + b_out   (in-place, no activation)
// ---------------------------------------------------------------------------
__global__ __launch_bounds__(256)
void output_layer_kernel(float* __restrict__ h,
                         const float* __restrict__ W,    // [128,128]
                         const float* __restrict__ bias)
{
  const int LD = 132;
  __shared__ float fs[16 * LD];
  const int tid = threadIdx.x;
  const int lane = tid & 31, wave = tid >> 5;
  const long base = (long)blockIdx.x * 16;

#ifdef HAVE_ASYNC_LDS
  {
    int r0 = wave * 2;
    #pragma unroll
    for (int i = 0; i < 2; ++i) {
      int r = r0 + i;
      async_copy16(h + (base + r) * H_ + lane * 4, &fs[r * LD + lane * 4]);
    }
    wait_async0();
  }
#else
  for (int idx = tid; idx < 16 * H_; idx += 256) {
    int r = idx >> 7, c = idx & 127;
    fs[r * LD + c] = h[(base + r) * H_ + c];
  }
#endif
  __syncthreads();

  const int col = lane & 15, khalf = (lane >> 4) * 2, n0 = wave * 16;

  v8f acc;
  const float bv = bias[n0 + col];
  #pragma unroll
  for (int j = 0; j < 8; ++j) acc[j] = bv;
  #pragma unroll
  for (int s = 0; s < 32; ++s) {
    int k0 = 4 * s + khalf;
    v2f a; a.x = fs[col * LD + k0]; a.y = fs[col * LD + k0 + 1];
    v2f b; b.x = W[(k0) * H_ + n0 + col]; b.y = W[(k0 + 1) * H_ + n0 + col];
    acc = wmma_f32(a, b, acc);
  }
  const int rbase = (lane >> 4) * 8;
  #pragma unroll
  for (int j = 0; j < 8; ++j)
    h[(base + rbase + j) * H_ + n0 + col] = acc[j];
}

// ---------------------------------------------------------------------------
// Kernel 5: affine coupling head (PHASE=0): both H+3 -> H hiddens via WMMA,
// H -> 3 heads via VALU dots, fused tanh/exp/mask, per-batch log-det.
// ---------------------------------------------------------------------------
__global__ __launch_bounds__(256)
void coupling_kernel(const float* __restrict__ nf,      // node_features (= h)
                     const float* __restrict__ coords,
                     const unsigned char* __restrict__ maskel,
                     const float* __restrict__ Ws1, const float* __restrict__ bs1,
                     const float* __restrict__ Ws2, const float* __restrict__ bs2,
                     const float* __restrict__ Wh1, const float* __restrict__ bh1,
                     const float* __restrict__ Wh2, const float* __restrict__ bh2,
                     float* __restrict__ out, float* __restrict__ logdet)
{
  const int KP = 132, LDF = 133, LDH = 133;   // K=131 padded to 132
  __shared__ float feat[16 * LDF];
  __shared__ float hid [16 * LDH];
  __shared__ float cmaskf[16];
  __shared__ float coordt[48];
  __shared__ float lsbuf[48];
  const int tid = threadIdx.x;
  const long base = (long)blockIdx.x * 16;

  if (tid < 16) {
    long node = base + tid;
    int n = (int)(node & (N_ - 1));
    float valid = maskel[node] ? 0.f : 1.f;
    cmaskf[tid] = ((n & 1) == 0) ? valid : 0.f;   // PHASE == 0
  }
  if (tid < 48) coordt[tid] = coords[base * 3 + tid];
  __syncthreads();

  for (int idx = tid; idx < 16 * KP; idx += 256) {
    int r = idx / KP, c = idx % KP;
    float v = 0.f;
    if (c < H_)          v = nf[(base + r) * H_ + c];
    else if (c < H_ + 3) v = coordt[r * 3 + (c - H_)] * (1.f - cmaskf[r]);
    feat[r * LDF + c] = v;
  }
  __syncthreads();

  const int lane = tid & 31, wave = tid >> 5;
  const int col = lane & 15, khalf = (lane >> 4) * 2, n0 = wave * 16;
  const int rbase = (lane >> 4) * 8;

  // ---- scale-net hidden: relu(feat @ Ws1 + bs1) ----
  v8f acc;
  float bv = bs1[n0 + col];
  #pragma unroll
  for (int j = 0; j < 8; ++j) acc[j] = bv;
  #pragma unroll
  for (int s = 0; s < 33; ++s) {
    int k0 = 4 * s + khalf;
    v2f a; a.x = feat[col * LDF + k0]; a.y = feat[col * LDF + k0 + 1];
    v2f b;
    b.x = (k0     < 131) ? Ws1[(k0    ) * H_ + n0 + col] : 0.f;
    b.y = (k0 + 1 < 131) ? Ws1[(k0 + 1) * H_ + n0 + col] : 0.f;
    acc = wmma_f32(a, b, acc);
  }
  #pragma unroll
  for (int j = 0; j < 8; ++j)
    hid[(rbase + j) * LDH + n0 + col] = fmaxf(acc[j], 0.f);
  __syncthreads();

  // ---- raw_scales = hid @ Ws2 + bs2 (3-wide head) ----
  float rawv = 0.f;
  if (tid < 48) {
    int m = tid / 3, d = tid % 3;
    float s = bs2[d];
    for (int j = 0; j < H_; ++j) s += hid[m * LDH + j] * Ws2[j * 3 + d];
    rawv = s;
  }
  __syncthreads();

  // ---- shift-net hidden: relu(feat @ Wh1 + bh1) ----
  bv = bh1[n0 + col];
  #pragma unroll
  for (int j = 0; j < 8; ++j) acc[j] = bv;
  #pragma unroll
  for (int s = 0; s < 33; ++s) {
    int k0 = 4 * s + khalf;
    v2f a; a.x = feat[col * LDF + k0]; a.y = feat[col * LDF + k0 + 1];
    v2f b;
    b.x = (k0     < 131) ? Wh1[(k0    ) * H_ + n0 + col] : 0.f;
    b.y = (k0 + 1 < 131) ? Wh1[(k0 + 1) * H_ + n0 + col] : 0.f;
    acc = wmma_f32(a, b, acc);
  }
  #pragma unroll
  for (int j = 0; j < 8; ++j)
    hid[(rbase + j) * LDH + n0 + col] = fmaxf(acc[j], 0.f);
  __syncthreads();

  // ---- shifts + fused elementwise + log-det ----
  if (tid < 48) {
    int m = tid / 3, d = tid % 3;
    float sh = bh2[d];
    for (int j = 0; j < H_; ++j) sh += hid[m * LDH + j] * Wh2[j * 3 + d];
    float cm = cmaskf[m];
    float ls = tanhf(rawv) * 2.f * cm;        // cm==0 -> ls==0 -> exp==1
    float co = coordt[tid];
    out[base * 3 + tid] = expf(ls) * co + sh * cm;  // == where(mask, out, coords)
    lsbuf[tid] = ls;
  }
  __syncthreads();
  if (tid == 0) {
    float s = 0.f;
    for (int i = 0; i < 48; ++i) s += lsbuf[i];
    atomicAdd(&logdet[(int)(base >> 13)], s);  // batch = node / N_ (N_=8192)
  }
}

// ---------------------------------------------------------------------------
extern "C" void kernel_launch(void* const* d_in, const int* in_sizes, int n_in,
                              void* d_out, int out_size, void* d_ws, size_t ws_size,
                              hipStream_t stream) {
  const float*         coords = (const float*)d_in[0];
  const int*           atypes = (const int*)d_in[1];
  const int*           adj    = (const int*)d_in[2];
  const int*           ebi    = (const int*)d_in[3];
  const unsigned char* maskel = (const unsigned char*)d_in[4];
  const float* embed = (const float*)d_in[5];
  const float* W_in  = (const float*)d_in[6];
  const float* b_in  = (const float*)d_in[7];
  const float* msg_W = (const float*)d_in[8];
  const float* msg_b = (const float*)d_in[9];
  const float* upd_W = (const float*)d_in[10];
  const float* upd_b = (const float*)d_in[11];
  const float* W_out = (const float*)d_in[12];
  const float* b_out = (const float*)d_in[13];
  const float* Ws1   = (const float*)d_in[14];
  const float* bs1   = (const float*)d_in[15];
  const float* Ws2   = (const float*)d_in[16];
  const float* bs2   = (const float*)d_in[17];
  const float* Wh1   = (const float*)d_in[18];
  const float* bh1   = (const float*)d_in[19];
  const float* Wh2   = (const float*)d_in[20];
  const float* bh2   = (const float*)d_in[21];

  const size_t BNH = (size_t)B_ * N_ * H_;
  float* h   = (float*)d_ws;        // 134 MB, becomes node_features in place
  float* agg = h + BNH;             // 134 MB
  float* outp   = (float*)d_out;                 // [B,N,3]
  float* logdet = outp + (size_t)B_ * N_ * 3;    // [B]

  const int node_tiles = (B_ * N_) / 16;         // 16384
  input_layer_kernel<<<node_tiles, 256, 0, stream>>>(coords, atypes, maskel,
                                                     embed, W_in, b_in, h);
  for (int l = 0; l < L_; ++l) {
    (void)hipMemsetAsync(agg, 0, BNH * sizeof(float), stream);
    message_kernel<<<8192, 256, 0, stream>>>(h, adj, ebi,
                                             msg_W + (size_t)l * H_ * H_,
                                             msg_b + (size_t)l * H_,
                                             agg, E_ / 16);
    update_kernel<<<node_tiles, 256, 0, stream>>>(h, agg,
                                                  upd_W + (size_t)l * 2 * H_ * H_,
                                                  upd_b + (size_t)l * H_,
                                                  maskel);
  }
  output_layer_kernel<<<node_tiles, 256, 0, stream>>>(h, W_out, b_out);

  (void)hipMemsetAsync(logdet, 0, B_ * sizeof(float), stream);
  coupling_kernel<<<node_tiles, 256, 0, stream>>>(h, coords, maskel,
                                                  Ws1, bs1, Ws2, bs2,
                                                  Wh1, bh1, Wh2, bh2,
                                                  outp, logdet);
}